// RNN_66133906424542
// MI455X (gfx1250) — compile-verified
//
#include <hip/hip_runtime.h>
#include <math.h>

#define T_SEQ 1024
#define BATCH 64
#define DIN   1024
#define HDIM  1024

typedef float v2f __attribute__((ext_vector_type(2)));
typedef float v8f __attribute__((ext_vector_type(8)));

#if defined(__HIP_DEVICE_COMPILE__) && defined(__AMDGCN__)
#if !__has_builtin(__builtin_amdgcn_wmma_f32_16x16x4_f32)
#error "device: __builtin_amdgcn_wmma_f32_16x16x4_f32 not available"
#endif
#endif

static __device__ __forceinline__ v8f wmma4(v2f a, v2f b, v8f c) {
  // D = A(16x4 f32) * B(4x16 f32) + C(16x16 f32)
  return __builtin_amdgcn_wmma_f32_16x16x4_f32(
      /*neg_a=*/false, a, /*neg_b=*/false, b,
      /*c_mod=*/(short)0, c, /*reuse_a=*/false, /*reuse_b=*/false);
}

// ---------------------------------------------------------------------------
// Phase 1: xproj = x @ W_ih^T + b_ih   (M=65536, N=1024, K=1024)
// Wave tile: 32(M) x 64(N) -> 2 A-frags x 4 B-frags, 8 accumulators.
// 8 WMMA per 6 b64 loads (10.7 FLOP/B from cache). 8 waves/block, 4096 blocks.
// ---------------------------------------------------------------------------
__global__ __launch_bounds__(256) void xproj_kernel(
    const float* __restrict__ x, const float* __restrict__ Wih,
    const float* __restrict__ bih, float* __restrict__ out)
{
  const int lane   = (int)(threadIdx.x & 31u);
  const int wave   = (int)(threadIdx.x >> 5u);
  const int laneM  = lane & 15;       // M (A) / N (B,C) index
  const int laneHi = lane >> 4;       // K-half (A,B) / M-half (C)

  const unsigned waveId = blockIdx.x * 8u + (unsigned)wave;  // 0..32767
  const unsigned m0 = (waveId >> 4) * 32u;   // 0..65504
  const unsigned n0 = (waveId & 15u) * 64u;  // 0..960

  const float* aP0 = x   + (size_t)(m0 +  0u + laneM) * DIN + 2u * laneHi;
  const float* aP1 = x   + (size_t)(m0 + 16u + laneM) * DIN + 2u * laneHi;
  const float* bP0 = Wih + (size_t)(n0 +  0u + laneM) * DIN + 2u * laneHi;
  const float* bP1 = Wih + (size_t)(n0 + 16u + laneM) * DIN + 2u * laneHi;
  const float* bP2 = Wih + (size_t)(n0 + 32u + laneM) * DIN + 2u * laneHi;
  const float* bP3 = Wih + (size_t)(n0 + 48u + laneM) * DIN + 2u * laneHi;

  v8f c00 = {}, c01 = {}, c02 = {}, c03 = {};
  v8f c10 = {}, c11 = {}, c12 = {}, c13 = {};

  #pragma unroll 4
  for (int kk = 0; kk < DIN / 4; ++kk) {
    v2f a0 = *(const v2f*)(aP0 + 4 * kk);
    v2f a1 = *(const v2f*)(aP1 + 4 * kk);
    v2f b0 = *(const v2f*)(bP0 + 4 * kk);
    v2f b1 = *(const v2f*)(bP1 + 4 * kk);
    v2f b2 = *(const v2f*)(bP2 + 4 * kk);
    v2f b3 = *(const v2f*)(bP3 + 4 * kk);
    c00 = wmma4(a0, b0, c00);  c01 = wmma4(a0, b1, c01);
    c02 = wmma4(a0, b2, c02);  c03 = wmma4(a0, b3, c03);
    c10 = wmma4(a1, b0, c10);  c11 = wmma4(a1, b1, c11);
    c12 = wmma4(a1, b2, c12);  c13 = wmma4(a1, b3, c13);
  }

  const float s0 = bih[n0 +  0u + laneM];
  const float s1 = bih[n0 + 16u + laneM];
  const float s2 = bih[n0 + 32u + laneM];
  const float s3 = bih[n0 + 48u + laneM];

  #pragma unroll
  for (int r = 0; r < 8; ++r) {
    const unsigned row0 = m0 + (unsigned)r + 8u * (unsigned)laneHi;
    const unsigned row1 = row0 + 16u;
    float* o0 = out + (size_t)row0 * HDIM + n0 + (unsigned)laneM;
    float* o1 = out + (size_t)row1 * HDIM + n0 + (unsigned)laneM;
    o0[ 0] = c00[r] + s0;  o0[16] = c01[r] + s1;
    o0[32] = c02[r] + s2;  o0[48] = c03[r] + s3;
    o1[ 0] = c10[r] + s0;  o1[16] = c11[r] + s1;
    o1[32] = c12[r] + s2;  o1[48] = c13[r] + s3;
  }
}

// ---------------------------------------------------------------------------
// Phase 2: persistent recurrence, 8-way split-K.
//   h_t = tanh(xproj_t + h_{t-1} @ W_hh^T + b_hh)
// 64 blocks (4 mTiles x 16 nGroups), 8 waves each. Block owns one 16x64 output
// tile; wave w accumulates K in [128w, 128w+128) (WMMA chain depth 32).
// Partials exchanged via LDS; waves 0..3 reduce subtile j, add xproj + bias,
// tanh, store. Device-scope atomic barrier between steps.
// ---------------------------------------------------------------------------
__global__ __launch_bounds__(256) void rnn_rec_kernel(
    const float* __restrict__ h0, const float* __restrict__ Whh,
    const float* __restrict__ bhh, float* __restrict__ out,
    float* __restrict__ hn, unsigned* __restrict__ counter)
{
  __shared__ float part[8 * 4 * 256];  // [kq][j][r*32+lane], 32 KB

  const int lane   = (int)(threadIdx.x & 31u);
  const int wave   = (int)(threadIdx.x >> 5u);   // = K-chunk id kq, 0..7
  const int laneM  = lane & 15;
  const int laneHi = lane >> 4;

  const unsigned m0 = (blockIdx.x >> 4) * 16u;   // batch rows  0..48
  const unsigned n0 = (blockIdx.x & 15u) * 64u;  // hidden cols 0..960
  const unsigned k0 = (unsigned)wave * 128u;     // this wave's K chunk

  // Loop-invariant W_hh fragment pointers (L2-resident across all steps).
  const float* bP0 = Whh + (size_t)(n0 +  0u + laneM) * HDIM + k0 + 2u * laneHi;
  const float* bP1 = Whh + (size_t)(n0 + 16u + laneM) * HDIM + k0 + 2u * laneHi;
  const float* bP2 = Whh + (size_t)(n0 + 32u + laneM) * HDIM + k0 + 2u * laneHi;
  const float* bP3 = Whh + (size_t)(n0 + 48u + laneM) * HDIM + k0 + 2u * laneHi;

  // Bias for the subtile this wave reduces (waves 0..3 only).
  const float sb = (wave < 4) ? bhh[n0 + 16u * (unsigned)wave + laneM] : 0.0f;

  const size_t BH = (size_t)BATCH * HDIM;

  for (int t = 0; t < T_SEQ; ++t) {
    const float* hsrc = (t == 0) ? h0 : (out + (size_t)(t - 1) * BH);
    const float* aPtr = hsrc + (size_t)(m0 + laneM) * HDIM + k0 + 2u * laneHi;

    v8f c0 = {}, c1 = {}, c2 = {}, c3 = {};
    #pragma unroll 8
    for (int kk = 0; kk < 32; ++kk) {       // 128 / 4
      v2f a  = *(const v2f*)(aPtr + 4 * kk);
      v2f b0 = *(const v2f*)(bP0  + 4 * kk);
      v2f b1 = *(const v2f*)(bP1  + 4 * kk);
      v2f b2 = *(const v2f*)(bP2  + 4 * kk);
      v2f b3 = *(const v2f*)(bP3  + 4 * kk);
      c0 = wmma4(a, b0, c0);
      c1 = wmma4(a, b1, c1);
      c2 = wmma4(a, b2, c2);
      c3 = wmma4(a, b3, c3);
    }

    // Publish partial C tiles to LDS: part[(kq*4+j)*256 + r*32 + lane]
    {
      const int base = wave * 4 * 256;
      #pragma unroll
      for (int r = 0; r < 8; ++r) {
        part[base +   0 + r * 32 + lane] = c0[r];
        part[base + 256 + r * 32 + lane] = c1[r];
        part[base + 512 + r * 32 + lane] = c2[r];
        part[base + 768 + r * 32 + lane] = c3[r];
      }
    }
    __syncthreads();

    if (wave < 4) {
      const int j = wave;                    // subtile (columns n0+16j..+15)
      float acc[8];
      #pragma unroll
      for (int r = 0; r < 8; ++r) acc[r] = 0.0f;
      #pragma unroll
      for (int kq = 0; kq < 8; ++kq) {
        const int base = (kq * 4 + j) * 256;
        #pragma unroll
        for (int r = 0; r < 8; ++r) acc[r] += part[base + r * 32 + lane];
      }
      float* obase = out + (size_t)t * BH;
      #pragma unroll
      for (int r = 0; r < 8; ++r) {
        const unsigned row = m0 + (unsigned)r + 8u * (unsigned)laneHi;
        const size_t idx = (size_t)row * HDIM + n0 + 16u * (unsigned)j
                           + (unsigned)laneM;
        const float v = tanhf(obase[idx] + acc[r] + sb);
        obase[idx] = v;
        if (t == T_SEQ - 1) hn[idx] = v;
      }
    }

    // ---- device-wide barrier (monotone counter, release/acquire) ----
    __threadfence();      // make every thread's global stores agent-visible
    __syncthreads();      // also protects LDS reuse next iteration
    if (threadIdx.x == 0) {
      __hip_atomic_fetch_add(counter, 1u, __ATOMIC_RELEASE,
                             __HIP_MEMORY_SCOPE_AGENT);
      const unsigned target = gridDim.x * (unsigned)(t + 1);
      while (__hip_atomic_load(counter, __ATOMIC_ACQUIRE,
                               __HIP_MEMORY_SCOPE_AGENT) < target) {
        __builtin_amdgcn_s_sleep(2);
      }
    }
    __syncthreads();
    __threadfence();      // acquire side: discard stale cached h_{t}
  }
}

__global__ void init_counter_kernel(unsigned* __restrict__ c) {
  if (threadIdx.x == 0 && blockIdx.x == 0) *c = 0u;
}

extern "C" void kernel_launch(void* const* d_in, const int* in_sizes, int n_in,
                              void* d_out, int out_size, void* d_ws, size_t ws_size,
                              hipStream_t stream) {
  (void)in_sizes; (void)n_in; (void)out_size; (void)ws_size;
  const float* x   = (const float*)d_in[0];  // (T, B, DIN)
  const float* h0  = (const float*)d_in[1];  // (1, B, H)
  const float* Wih = (const float*)d_in[2];  // (H, DIN)
  const float* Whh = (const float*)d_in[3];  // (H, H)
  const float* bih = (const float*)d_in[4];  // (H)
  const float* bhh = (const float*)d_in[5];  // (H)

  float* out = (float*)d_out;                        // (T,1,B,H) flat
  float* hn  = out + (size_t)T_SEQ * BATCH * HDIM;   // (1,1,B,H) flat
  unsigned* counter = (unsigned*)d_ws;

  init_counter_kernel<<<1, 64, 0, stream>>>(counter);

  // 2048 M-tiles(32) * 16 N-groups / 8 waves per block
  xproj_kernel<<<4096, 256, 0, stream>>>(x, Wih, bih, out);

  // 4 mTiles * 16 nGroups = 64 blocks, 8 waves (split-K) each
  rnn_rec_kernel<<<64, 256, 0, stream>>>(h0, Whh, bhh, out, hn, counter);
}